// NearestNeighborSampler_43928925503752
// MI455X (gfx1250) — compile-verified
//
#include <hip/hip_runtime.h>
#include <math.h>

// Problem constants (match reference setup_inputs()).
constexpr int B = 2048;     // batch rows
constexpr int F = 256;      // feature dim (K of the GEMM)
constexpr int Q = 32768;    // queue rows (N of the GEMM)
constexpr int P = 16;       // perplexity / top-k
constexpr int NPART = 16;               // column partitions
constexpr int PARTN = Q / NPART;        // 2048 columns per partition
constexpr int COLTILES = PARTN / 16;    // 128 16-col WMMA tiles per partition
constexpr int KSTEPS = F / 4;           // 64 chained 16x16x4 f32 WMMAs
constexpr int WAVES = 8;                // waves per workgroup
constexpr int ROWS_PER_WG = WAVES * 16; // 128 rows per workgroup
constexpr int LBS = 260;                // ldsB row stride (floats); stride%64==4 -> conflict-free

typedef __attribute__((ext_vector_type(2))) float v2f;
typedef __attribute__((ext_vector_type(8))) float v8f;

// ---------------------------------------------------------------------------
// Kernel 1: squared norms of every logical queue row.
// queue[j] = (j < B) ? x[j] : queue_buf[j - B];  note x2[i] == q2[i] for i < B.
// ---------------------------------------------------------------------------
__global__ void __launch_bounds__(256)
nns_norms(const float* __restrict__ x, const float* __restrict__ qb,
          float* __restrict__ q2) {
  int j = blockIdx.x * blockDim.x + threadIdx.x;
  if (j >= Q) return;
  const float* r = (j < B) ? (x + (size_t)j * F) : (qb + (size_t)(j - B) * F);
  float s = 0.0f;
#pragma unroll 8
  for (int k = 0; k < F; k += 4) {
    float4 v = *(const float4*)(r + k);
    s += v.x * v.x + v.y * v.y + v.z * v.z + v.w * v.w;
  }
  q2[j] = s;
}

// Issue one thread's share of a B-tile stage: 16 floats = 4 async b128 copies
// straight into LDS (no VGPR round trip), tracked on ASYNCcnt.
__device__ __forceinline__ void stage_tile_async(const float* __restrict__ x,
                                                 const float* __restrict__ qb,
                                                 unsigned ldsDstBytes,
                                                 int qrow, int sk) {
  const float* src =
      ((qrow < B) ? (x + (size_t)qrow * F) : (qb + (size_t)(qrow - B) * F)) + sk;
#pragma unroll
  for (int u = 0; u < 4; ++u) {
    asm volatile("global_load_async_to_lds_b128 %0, %1, off"
                 :: "v"(ldsDstBytes + 16u * (unsigned)u), "v"(src + 4 * u)
                 : "memory");
  }
}

// ---------------------------------------------------------------------------
// Kernel 2: fused fp32-WMMA GEMM + distance + per-partition ordered top-16.
// 8 waves per block; wave w owns rows [128*bx + 16*w, +16). The 16-col x K=256
// B tile is async-staged into a double-buffered LDS tile once per block and
// consumed by all 8 waves via conflict-free ds_load_b64 (8x less L2 traffic).
// A fragments (16 rows x K=256 fp32) stay register-resident per wave.
// ---------------------------------------------------------------------------
__global__ void __launch_bounds__(256)
nns_gemm_topk(const float* __restrict__ x, const float* __restrict__ qb,
              const float* __restrict__ q2,
              float* __restrict__ pdist, int* __restrict__ pidx) {
  const int tid  = threadIdx.x;          // 0..255
  const int lane = tid & 31;
  const int wave = tid >> 5;
  const int rowBase = blockIdx.x * ROWS_PER_WG + wave * 16;
  const int part = blockIdx.y;
  const int colPartBase = part * PARTN;
  const int hi = lane >> 4;              // 0 = lanes 0-15, 1 = lanes 16-31
  const int l16 = lane & 15;
  const int koff = hi * 2;               // upper half-wave holds K+2, K+3

  __shared__ float ldsB[2][16 * LBS];    // double-buffered B tile [col][k]
  __shared__ float ldsD[WAVES][16][17];  // per-wave distance transpose pad

  const unsigned ldsB0 = (unsigned)(uintptr_t)&ldsB[0][0];
  const unsigned ldsB1 = (unsigned)(uintptr_t)&ldsB[1][0];

  // staging assignment: thread t stages col (t&15), k-range [(t>>4)*16, +16)
  const int scol = tid & 15;
  const int sk = (tid >> 4) * 16;
  const unsigned myDstOff = (unsigned)((scol * LBS + sk) * 4);

  // --- A fragments: per 16x16x4 WMMA, lane holds {A[M][k], A[M][k+1]}
  //     (lanes 0-15: M=lane, K=k..k+1; lanes 16-31: M=lane-16, K=k+2..k+3)
  const float* aptr = x + (size_t)(rowBase + l16) * F + koff;
  v2f a[KSTEPS];
#pragma unroll
  for (int s = 0; s < KSTEPS; ++s)
    a[s] = *(const v2f*)(aptr + 4 * s);

  // ||x_row||^2 for this half-wave's 8 output rows (x2[i] == q2[i], i < B)
  float x2v[8];
#pragma unroll
  for (int v = 0; v < 8; ++v)
    x2v[v] = q2[rowBase + hi * 8 + v];

  // sorted ascending top-16 (meaningful on lanes 0-15; lane r owns row r)
  float val[P];
  int   idx[P];
#pragma unroll
  for (int t = 0; t < P; ++t) { val[t] = 3.0e38f; idx[t] = -1; }

  // prologue: stage tile 0 into buffer 0
  stage_tile_async(x, qb, ldsB0 + myDstOff, colPartBase + scol, sk);
  asm volatile("s_wait_asynccnt 0x0" ::: "memory");
  __syncthreads();

  for (int ct = 0; ct < COLTILES; ++ct) {
    const int cur = ct & 1;
    // overlap: async-stage tile ct+1 into the other buffer while computing
    if (ct + 1 < COLTILES) {
      const int ncol0 = colPartBase + (ct + 1) * 16;
      stage_tile_async(x, qb, (cur ? ldsB0 : ldsB1) + myDstOff, ncol0 + scol, sk);
    }

    const int col0 = colPartBase + ct * 16;
    const int qrow = col0 + l16;         // queue row backing column N=l16

    // --- chained fp32 WMMA over K: D += A(16x4) * B(4x16), B from LDS
    const float* bbase = &ldsB[cur][l16 * LBS + koff];
    v8f c = {};
#pragma unroll
    for (int s = 0; s < KSTEPS; ++s) {
      v2f b = *(const v2f*)(bbase + 4 * s);   // ds_load_b64, conflict-free
      c = __builtin_amdgcn_wmma_f32_16x16x4_f32(
          /*neg_a=*/false, a[s], /*neg_b=*/false, b,
          /*c_mod=*/(short)0, c, /*reuse_a=*/false, /*reuse_b=*/false);
    }

    // --- dot products -> euclidean distances, diagonal masked
    const float q2c = q2[qrow];
#pragma unroll
    for (int v = 0; v < 8; ++v) {
      const int m = hi * 8 + v;          // output row within tile
      const int grow = rowBase + m;      // global row index
      float d2 = x2v[v] + q2c - 2.0f * c[v];
      float dist = sqrtf(fmaxf(d2, 1e-12f));
      if (qrow == grow) dist = 1.0e30f;  // self-match mask
      ldsD[wave][m][l16] = dist;
    }

    // --- lanes 0-15: scan this tile's 16 columns for row `lane`,
    //     stable sorted insertion (strict < keeps ties index-ascending,
    //     matching jax.lax.top_k ordering). Same-wave LDS: no barrier needed.
    if (lane < 16) {
      for (int n = 0; n < 16; ++n) {
        float d = ldsD[wave][lane][n];
        if (d < val[P - 1]) {
          val[P - 1] = d;
          idx[P - 1] = col0 + n;
#pragma unroll
          for (int t = P - 1; t >= 1; --t) {
            if (val[t] < val[t - 1]) {
              float tv = val[t]; val[t] = val[t - 1]; val[t - 1] = tv;
              int   ti = idx[t]; idx[t] = idx[t - 1]; idx[t - 1] = ti;
            }
          }
        }
      }
    }

    // next tile fully staged + all waves done reading current buffer
    asm volatile("s_wait_asynccnt 0x0" ::: "memory");
    __syncthreads();
  }

  // --- write this partition's sorted top-16 for each of the 16 rows
  if (lane < 16) {
    const int row = rowBase + lane;
    float* pd = pdist + ((size_t)row * NPART + part) * P;
    int*   pi = pidx  + ((size_t)row * NPART + part) * P;
#pragma unroll
    for (int t = 0; t < P; ++t) { pd[t] = val[t]; pi[t] = idx[t]; }
  }
}

// ---------------------------------------------------------------------------
// Kernel 3: merge the 16 partial top-16 lists per row (partition-major order
// preserves tie stability), select entry jdx[row], gather the queue row.
// ---------------------------------------------------------------------------
__global__ void __launch_bounds__(256)
nns_merge_gather(const float* __restrict__ x, const float* __restrict__ qb,
                 const int* __restrict__ jdx,
                 const float* __restrict__ pdist, const int* __restrict__ pidx,
                 float* __restrict__ out) {
  const int row = blockIdx.x;
  __shared__ int ssel;

  if (threadIdx.x == 0) {
    float val[P];
    int   idx[P];
#pragma unroll
    for (int t = 0; t < P; ++t) { val[t] = 3.0e38f; idx[t] = -1; }
    const float* pd = pdist + (size_t)row * NPART * P;
    const int*   pi = pidx  + (size_t)row * NPART * P;
    for (int u = 0; u < NPART * P; ++u) {
      float d = pd[u];
      if (d < val[P - 1]) {
        val[P - 1] = d;
        idx[P - 1] = pi[u];
#pragma unroll
        for (int t = P - 1; t >= 1; --t) {
          if (val[t] < val[t - 1]) {
            float tv = val[t]; val[t] = val[t - 1]; val[t - 1] = tv;
            int   ti = idx[t]; idx[t] = idx[t - 1]; idx[t - 1] = ti;
          }
        }
      }
    }
    const int k = jdx[row];
    int s = idx[0];
#pragma unroll
    for (int t = 1; t < P; ++t)
      if (t == k) s = idx[t];
    ssel = s;
  }
  __syncthreads();

  const int s = ssel;
  out[(size_t)row * F + threadIdx.x] =
      (s < B) ? x[(size_t)s * F + threadIdx.x]
              : qb[(size_t)(s - B) * F + threadIdx.x];
}

// ---------------------------------------------------------------------------
extern "C" void kernel_launch(void* const* d_in, const int* in_sizes, int n_in,
                              void* d_out, int out_size, void* d_ws, size_t ws_size,
                              hipStream_t stream) {
  const float* x   = (const float*)d_in[0];   // [B, F]
  const float* qb  = (const float*)d_in[1];   // [Q, F]
  const int*   jdx = (const int*)d_in[2];     // [B]
  // d_in[3] = perplexity scalar (hardcoded P=16)
  float* out = (float*)d_out;                 // [B, F]

  char* ws = (char*)d_ws;
  float* q2    = (float*)ws;                                            // Q floats
  float* pdist = (float*)(ws + sizeof(float) * (size_t)Q);              // B*NPART*P floats
  int*   pidx  = (int*)(ws + sizeof(float) * (size_t)Q
                           + sizeof(float) * (size_t)B * NPART * P);    // B*NPART*P ints

  nns_norms<<<Q / 256, 256, 0, stream>>>(x, qb, q2);
  nns_gemm_topk<<<dim3(B / ROWS_PER_WG, NPART), 256, 0, stream>>>(x, qb, q2, pdist, pidx);
  nns_merge_gather<<<B, F, 0, stream>>>(x, qb, jdx, pdist, pidx, out);
}